// NodeClassifier_56727928045714
// MI455X (gfx1250) — compile-verified
//
#include <hip/hip_runtime.h>
#include <math.h>

typedef float v2f __attribute__((ext_vector_type(2)));
typedef float v8f __attribute__((ext_vector_type(8)));

#define KK 3
#define DD 512
#define CC 16
#define NEG_SLOPE 0.2f

// Order-preserving f32 <-> u32 key (for atomicMax over signed floats).
__device__ __forceinline__ unsigned f32_key(float f) {
    unsigned b = __float_as_uint(f);
    return (b & 0x80000000u) ? ~b : (b | 0x80000000u);
}
__device__ __forceinline__ float key_f32(unsigned k) {
    unsigned b = (k & 0x80000000u) ? (k ^ 0x80000000u) : ~k;
    return __uint_as_float(b);
}

__global__ void zero_kernel(unsigned* __restrict__ mkey, float* __restrict__ den, int n) {
    int i = blockIdx.x * blockDim.x + threadIdx.x;
    if (i < n) { mkey[i] = 0u; den[i] = 0.0f; }
}

// feat = h @ W, res = h @ res_W  (fused: A tile of h loaded once, two WMMA chains).
// W / res_W staged in LDS once per workgroup; inner-loop B fetches ride the DS pipe
// so the VMEM pipe is dedicated to streaming h (the HBM-bound operand).
__global__ __launch_bounds__(128) void gat_gemm_kernel(
    const float* __restrict__ h, const float* __restrict__ W,
    const float* __restrict__ rW, float* __restrict__ feat,
    float* __restrict__ out, int N)
{
    __shared__ __align__(16) float ldsW[DD * CC];   // 32 KB
    __shared__ __align__(16) float ldsR[DD * CC];   // 32 KB

    const int k = blockIdx.y;
    const float* Wk  = W  + (long)k * DD * CC;
    const float* rWk = rW + (long)k * DD * CC;

    // Cooperative fill: 128 threads x float4, 16 passes per matrix.
    for (int i = threadIdx.x * 4; i < DD * CC; i += 128 * 4) {
        *(float4*)(ldsW + i) = *(const float4*)(Wk + i);
        *(float4*)(ldsR + i) = *(const float4*)(rWk + i);
    }
    __syncthreads();

    const int lane = threadIdx.x & 31;
    const int wave = threadIdx.x >> 5;
    const int tile = blockIdx.x * 4 + wave;
    const int ntiles = N >> 4;           // N divisible by 16 (50000)
    if (tile >= ntiles) return;          // uniform per wave -> EXEC stays all-ones

    const float* hk = h + (long)k * N * DD;

    // ISA 32-bit A 16x4 layout: lanes 0-15 hold K={0,1}, lanes 16-31 hold K={2,3}; M = lane&15.
    const int M  = lane & 15;
    const int K0 = (lane >> 4) << 1;
    const long rowoff = (long)(tile * 16 + M) * DD;

    v8f accF = {};
    v8f accR = {};
#pragma unroll 4
    for (int kk = 0; kk < DD; kk += 4) {
        v2f a = *(const v2f*)(hk + rowoff + kk + K0);
        v2f bW, bR;                       // two elements 64 B apart -> ds_load_2addr
        bW.x = ldsW[(kk + K0)     * CC + M];
        bW.y = ldsW[(kk + K0 + 1) * CC + M];
        bR.x = ldsR[(kk + K0)     * CC + M];
        bR.y = ldsR[(kk + K0 + 1) * CC + M];
        accF = __builtin_amdgcn_wmma_f32_16x16x4_f32(false, a, false, bW, (short)0, accF, false, false);
        accR = __builtin_amdgcn_wmma_f32_16x16x4_f32(false, a, false, bR, (short)0, accR, false, false);
    }

    // C/D layout: VGPR r -> row r (lanes 0-15) / row r+8 (lanes 16-31), col = lane&15.
    float* featk = feat + (long)k * N * CC;
    float* outk  = out  + (long)k * N * CC;
    const int rbase = tile * 16 + ((lane >> 4) << 3);
    const int col   = lane & 15;
#pragma unroll
    for (int r = 0; r < 8; ++r) {
        long off = (long)(rbase + r) * CC + col;
        featk[off] = accF[r];
        outk[off]  = accR[r];   // residual term: out starts as h @ res_W
    }
}

// el[n] = dot(feat[n], attn_l), er[n] = dot(feat[n], attn_r)
__global__ void attn_node_kernel(const float* __restrict__ feat,
                                 const float* __restrict__ al,
                                 const float* __restrict__ ar,
                                 float* __restrict__ el, float* __restrict__ er, int N)
{
    int n = blockIdx.x * blockDim.x + threadIdx.x;
    int k = blockIdx.y;
    if (n >= N) return;
    const float* f  = feat + ((long)k * N + n) * CC;
    const float* vl = al + k * CC;
    const float* vr = ar + k * CC;
    float sl = 0.0f, sr = 0.0f;
#pragma unroll
    for (int c = 0; c < CC; ++c) {
        float v = f[c];
        sl += v * vl[c];
        sr += v * vr[c];
    }
    el[(long)k * N + n] = sl;
    er[(long)k * N + n] = sr;
}

// score = leaky_relu(el[src]+er[dst]); segment-max via ordered-key atomicMax.
// Edge list read once; K=3 handled in-thread (el/er/mkey are L2-resident).
__global__ void edge_max_kernel(const int* __restrict__ src, const int* __restrict__ dst,
                                const float* __restrict__ el, const float* __restrict__ er,
                                float* __restrict__ score, unsigned* __restrict__ mkey,
                                int N, int E)
{
    long e = (long)blockIdx.x * blockDim.x + threadIdx.x;
    if (e >= E) return;
    int s = src[e], d = dst[e];
#pragma unroll
    for (int k = 0; k < KK; ++k) {
        float v = el[(long)k * N + s] + er[(long)k * N + d];
        v = v > 0.0f ? v : NEG_SLOPE * v;
        score[(long)k * E + e] = v;
        atomicMax(mkey + (long)k * N + d, f32_key(v));
    }
}

// decode max keys; nodes w/o incoming edges (key==0 -> NaN) become 0 (matches reference)
__global__ void decode_m_kernel(const unsigned* __restrict__ mkey, float* __restrict__ mval, int n) {
    int i = blockIdx.x * blockDim.x + threadIdx.x;
    if (i >= n) return;
    float m = key_f32(mkey[i]);
    if (!(fabsf(m) < __builtin_inff())) m = 0.0f;  // NaN or +/-inf -> 0
    mval[i] = m;
}

// den[dst] += exp(score - m[dst])
__global__ void edge_sum_kernel(const int* __restrict__ dst, const float* __restrict__ score,
                                const float* __restrict__ mval, float* __restrict__ den,
                                int N, int E)
{
    long e = (long)blockIdx.x * blockDim.x + threadIdx.x;
    if (e >= E) return;
    int d = dst[e];
#pragma unroll
    for (int k = 0; k < KK; ++k) {
        long kd = (long)k * N + d;
        float ex = expf(score[(long)k * E + e] - mval[kd]);
        atomicAdd(den + kd, ex);
    }
}

// out[dst] += feat[src] * a ; 16 lanes per edge, one channel each (coalesced
// 64 B feat gather + coalesced float atomics into out).
__global__ void edge_scatter_kernel(const int* __restrict__ src, const int* __restrict__ dst,
                                    const float* __restrict__ score, const float* __restrict__ mval,
                                    const float* __restrict__ den, const float* __restrict__ feat,
                                    float* __restrict__ out, int N, int E)
{
    long t = (long)blockIdx.x * blockDim.x + threadIdx.x;
    long e = t >> 4;
    int  c = (int)(t & 15);
    if (e >= E) return;
    int s = src[e], d = dst[e];
#pragma unroll
    for (int k = 0; k < KK; ++k) {
        long kd = (long)k * N + d;
        float a = expf(score[(long)k * E + e] - mval[kd]) / fmaxf(den[kd], 1e-9f);
        float msg = feat[((long)k * N + s) * CC + c] * a;
        atomicAdd(out + kd * CC + c, msg);
    }
}

extern "C" void kernel_launch(void* const* d_in, const int* in_sizes, int n_in,
                              void* d_out, int out_size, void* d_ws, size_t ws_size,
                              hipStream_t stream) {
    const float* h   = (const float*)d_in[0];
    const float* W   = (const float*)d_in[1];
    const float* al  = (const float*)d_in[2];
    const float* ar  = (const float*)d_in[3];
    const float* rW  = (const float*)d_in[4];
    const int*   src = (const int*)d_in[5];
    const int*   dst = (const int*)d_in[6];
    const int E = in_sizes[5];
    const int N = in_sizes[0] / (KK * DD);
    float* out = (float*)d_out;

    // workspace layout (all f32-sized slots)
    float*    feat  = (float*)d_ws;                       // K*N*16
    float*    el    = feat + (long)KK * N * CC;           // K*N
    float*    er    = el   + (long)KK * N;                // K*N
    unsigned* mkey  = (unsigned*)(er + (long)KK * N);     // K*N
    float*    mval  = (float*)(mkey + (long)KK * N);      // K*N
    float*    den   = mval + (long)KK * N;                // K*N
    float*    score = den  + (long)KK * N;                // K*E

    const int nk = KK * N;
    zero_kernel<<<(nk + 255) / 256, 256, 0, stream>>>(mkey, den, nk);

    dim3 ggrid((N / 16 + 3) / 4, KK);
    gat_gemm_kernel<<<ggrid, 128, 0, stream>>>(h, W, rW, feat, out, N);

    attn_node_kernel<<<dim3((N + 255) / 256, KK), 256, 0, stream>>>(feat, al, ar, el, er, N);

    edge_max_kernel<<<(E + 255) / 256, 256, 0, stream>>>(src, dst, el, er, score, mkey, N, E);

    decode_m_kernel<<<(nk + 255) / 256, 256, 0, stream>>>(mkey, mval, nk);

    edge_sum_kernel<<<(E + 255) / 256, 256, 0, stream>>>(dst, score, mval, den, N, E);

    long scatter_threads = (long)E * 16;
    edge_scatter_kernel<<<(unsigned)((scatter_threads + 255) / 256), 256, 0, stream>>>(
        src, dst, score, mval, den, feat, out, N, E);
}